// MatchingNetworks_53515292508762
// MI455X (gfx1250) — compile-verified
//
#include <hip/hip_runtime.h>
#include <math.h>

// CDNA5 / gfx1250 wave32 WMMA types
typedef __attribute__((ext_vector_type(2))) float v2f;
typedef __attribute__((ext_vector_type(8))) float v8f;

// async global->LDS b64 builtin operates on 2 x i32 vectors
typedef __attribute__((ext_vector_type(2))) int v2i;
typedef __attribute__((address_space(1))) v2i gbl_b64;
typedef __attribute__((address_space(3))) v2i lds_b64;

#define EMB_D        64      // embedding dim (in_sizes[6] == 64 for this problem)
#define KB           48      // K-block; 21168 = 48 * 441
#define GEMM_THREADS 128     // 4 waves: wave w handles output columns [16w, 16w+16)
#define NCHUNK       3       // (16*KB/2) b64 chunks / 128 threads = 3 per thread
#define EPS          1e-8f

// ---------------------------------------------------------------------------
// C[M, 64] = X[M, Din] @ W[Din, 64] + bias   via V_WMMA_F32_16X16X4_F32
// One block: 16 rows of C, all 64 columns (wave = column tile).
// A-slabs (16xKB) double-buffered in LDS, filled with GLOBAL_LOAD_ASYNC_TO_LDS
// (ASYNCcnt-tracked) so HBM streaming overlaps the WMMA pipe.
// ---------------------------------------------------------------------------
__global__ __launch_bounds__(GEMM_THREADS)
void mn_embed_gemm(const float* __restrict__ X, const float* __restrict__ W,
                   const float* __restrict__ bias, float* __restrict__ C,
                   int M, int Din)
{
    __shared__ float a_sh[2][16][KB + 2];   // stride 50 floats: 8B-aligned, bank-spread

    const int tid  = threadIdx.x;
    const int wave = tid >> 5;              // 0..3 -> N tile
    const int lane = tid & 31;
    const int m0   = blockIdx.x * 16;
    const int n0   = wave * 16;

    // WMMA f32 16x16x4 operand layout (ISA 7.12.2):
    //  A: lanes 0-15 -> M=lane, {K0,K1}; lanes 16-31 -> M=lane-16, {K2,K3}
    //  B: lanes 0-15 -> N=lane, rows {K0,K1}; lanes 16-31 -> N=lane-16, rows {K2,K3}
    const int mA   = lane & 15;
    const int kSel = (lane >> 4) << 1;      // 0 or 2
    const int nc   = n0 + (lane & 15);      // this lane's output column

    // Per-thread async-copy chunk coordinates for a 16 x KB slab (b64 chunks)
    int          crow[NCHUNK];
    int          ccol[NCHUNK];
    const float* gsrc[NCHUNK];
    #pragma unroll
    for (int i = 0; i < NCHUNK; ++i) {
        int idx = tid + i * GEMM_THREADS;   // 0 .. 16*KB/2 - 1
        int r   = idx / (KB / 2);
        int c2  = idx % (KB / 2);
        int gr  = m0 + r;
        if (gr >= M) gr = M - 1;            // clamp (edge tile of support set)
        crow[i] = r;
        ccol[i] = 2 * c2;
        gsrc[i] = X + (size_t)gr * Din + 2 * c2;
    }

    const int nblk = Din / KB;              // 441 (exact)

    // Prologue: fill buffer 0 asynchronously
    #pragma unroll
    for (int i = 0; i < NCHUNK; ++i)
        __builtin_amdgcn_global_load_async_to_lds_b64(
            (gbl_b64*)gsrc[i],
            (lds_b64*)&a_sh[0][crow[i]][ccol[i]], 0, 0);

    v8f acc = {};

    for (int kb = 0; kb < nblk; ++kb) {
        const int cur = kb & 1;
        const int k0  = kb * KB;

        if (kb + 1 < nblk) {
            // Prefetch next slab into the other buffer, then wait for the
            // current slab only (async loads complete in issue order).
            #pragma unroll
            for (int i = 0; i < NCHUNK; ++i)
                __builtin_amdgcn_global_load_async_to_lds_b64(
                    (gbl_b64*)(gsrc[i] + (k0 + KB)),
                    (lds_b64*)&a_sh[cur ^ 1][crow[i]][ccol[i]], 0, 0);
            __builtin_amdgcn_s_wait_asynccnt(NCHUNK);
        } else {
            __builtin_amdgcn_s_wait_asynccnt(0);
        }
        __syncthreads();                    // all waves' slab writes visible

        #pragma unroll
        for (int kk = 0; kk < KB; kk += 4) {
            v2f a = *reinterpret_cast<const v2f*>(&a_sh[cur][mA][kk + kSel]);
            const float* wp = W + (size_t)(k0 + kk + kSel) * EMB_D + nc;
            v2f b;
            b.x = wp[0];
            b.y = wp[EMB_D];
            // (neg_a, A, neg_b, B, c_mod, C, reuse_a, reuse_b)
            acc = __builtin_amdgcn_wmma_f32_16x16x4_f32(
                false, a, false, b, (short)0, acc, false, false);
        }
        __syncthreads();                    // protect cur before it's refilled
    }

    // C/D layout: VGPR i -> M = i + 8*(lane>=16), N = lane&15
    const float bv    = bias[nc];
    const int   mBase = m0 + ((lane >> 4) << 3);
    #pragma unroll
    for (int i = 0; i < 8; ++i) {
        int r = mBase + i;
        if (r < M) C[(size_t)r * EMB_D + nc] = acc[i] + bv;
    }
}

// ---------------------------------------------------------------------------
// out[row] = emb[row] / max(||emb[row]||, eps)   (support set, 64 threads/row)
// ---------------------------------------------------------------------------
__global__ __launch_bounds__(EMB_D)
void mn_normalize_rows(const float* __restrict__ emb, float* __restrict__ out)
{
    __shared__ float red[EMB_D];
    const int t   = threadIdx.x;
    const int row = blockIdx.x;

    float v = emb[(size_t)row * EMB_D + t];
    red[t] = v * v;
    __syncthreads();
    if (t < 32) red[t] += red[t + 32];
    __syncthreads();
    if (t < 16) red[t] += red[t + 16];
    __syncthreads();
    if (t < 8)  red[t] += red[t + 8];
    __syncthreads();
    if (t < 4)  red[t] += red[t + 4];
    __syncthreads();
    if (t < 2)  red[t] += red[t + 2];
    __syncthreads();
    if (t == 0) red[0] = 1.0f / fmaxf(sqrtf(red[0] + red[1]), EPS);
    __syncthreads();
    out[(size_t)row * EMB_D + t] = v * red[0];
}

// ---------------------------------------------------------------------------
// Per query: cosine sims vs S normalized supports, softmax over S,
// scatter-add by label into n_way logits. One block (128 threads) per query.
// ---------------------------------------------------------------------------
__global__ __launch_bounds__(128)
void mn_attn_logits(const float* __restrict__ q_emb, const float* __restrict__ s_norm,
                    const int* __restrict__ labels, float* __restrict__ out,
                    int S, int n_way)
{
    __shared__ float q_sh[EMB_D];
    __shared__ float red[128];
    __shared__ float lg[32];

    const int t = threadIdx.x;
    const int q = blockIdx.x;

    // ---- query norm ----
    float v = 0.0f;
    if (t < EMB_D) { v = q_emb[(size_t)q * EMB_D + t]; q_sh[t] = v; }
    red[t] = v * v;
    __syncthreads();
    if (t < 64) red[t] += red[t + 64];
    __syncthreads();
    if (t < 32) red[t] += red[t + 32];
    __syncthreads();
    if (t < 16) red[t] += red[t + 16];
    __syncthreads();
    if (t < 8)  red[t] += red[t + 8];
    __syncthreads();
    if (t < 4)  red[t] += red[t + 4];
    __syncthreads();
    if (t < 2)  red[t] += red[t + 2];
    __syncthreads();
    if (t == 0) red[0] = 1.0f / fmaxf(sqrtf(red[0] + red[1]), EPS);
    __syncthreads();
    const float inv_qn = red[0];
    __syncthreads();

    // ---- cosine sims (supports already normalized, L2-resident) ----
    float sim = -3.0e38f;
    if (t < S) {
        const float* sp = s_norm + (size_t)t * EMB_D;
        float dot = 0.0f;
        #pragma unroll
        for (int d = 0; d < EMB_D; ++d) dot = fmaf(q_sh[d], sp[d], dot);
        sim = dot * inv_qn;
    }

    // ---- softmax: max ----
    red[t] = sim;
    __syncthreads();
    if (t < 64) red[t] = fmaxf(red[t], red[t + 64]);
    __syncthreads();
    if (t < 32) red[t] = fmaxf(red[t], red[t + 32]);
    __syncthreads();
    if (t < 16) red[t] = fmaxf(red[t], red[t + 16]);
    __syncthreads();
    if (t < 8)  red[t] = fmaxf(red[t], red[t + 8]);
    __syncthreads();
    if (t < 4)  red[t] = fmaxf(red[t], red[t + 4]);
    __syncthreads();
    if (t < 2)  red[t] = fmaxf(red[t], red[t + 2]);
    __syncthreads();
    if (t == 0) red[0] = fmaxf(red[0], red[1]);
    __syncthreads();
    const float mx = red[0];
    __syncthreads();

    // ---- softmax: exp + sum ----
    float e = (t < S) ? __expf(sim - mx) : 0.0f;
    red[t] = e;
    __syncthreads();
    if (t < 64) red[t] += red[t + 64];
    __syncthreads();
    if (t < 32) red[t] += red[t + 32];
    __syncthreads();
    if (t < 16) red[t] += red[t + 16];
    __syncthreads();
    if (t < 8)  red[t] += red[t + 8];
    __syncthreads();
    if (t < 4)  red[t] += red[t + 4];
    __syncthreads();
    if (t < 2)  red[t] += red[t + 2];
    __syncthreads();
    if (t == 0) red[0] += red[1];
    __syncthreads();
    const float inv_sum = 1.0f / red[0];
    __syncthreads();

    // ---- label scatter (attn @ one_hot) ----
    if (t < 32) lg[t] = 0.0f;
    __syncthreads();
    if (t < S) atomicAdd(&lg[labels[t]], e * inv_sum);
    __syncthreads();
    if (t < n_way) out[(size_t)q * n_way + t] = lg[t];
}

// ---------------------------------------------------------------------------
extern "C" void kernel_launch(void* const* d_in, const int* in_sizes, int n_in,
                              void* d_out, int out_size, void* d_ws, size_t ws_size,
                              hipStream_t stream)
{
    // setup_inputs order: support_images, support_labels, query_images,
    //                     n_way, k_shot, backbone_w, backbone_b
    const float* support_images = (const float*)d_in[0];
    const int*   support_labels = (const int*)d_in[1];
    const float* query_images   = (const float*)d_in[2];
    const float* backbone_w     = (const float*)d_in[5];
    const float* backbone_b     = (const float*)d_in[6];
    float*       out            = (float*)d_out;

    const int D     = in_sizes[6];          // 64 (kernels assume EMB_D == 64)
    const int Din   = in_sizes[5] / D;      // 21168 (multiple of KB=48)
    const int S     = in_sizes[0] / Din;    // 100
    const int Qn    = in_sizes[2] / Din;    // 4096
    const int n_way = out_size / Qn;        // 20
    (void)n_in; (void)ws_size; (void)D;

    // Workspace layout (floats): q_emb[Qn*64] | s_emb[Spad*64] | s_norm[Spad*64]
    const int Spad   = ((S + 15) / 16) * 16;
    float*    q_emb  = (float*)d_ws;
    float*    s_emb  = q_emb + (size_t)Qn * EMB_D;
    float*    s_norm = s_emb + (size_t)Spad * EMB_D;

    // 1) Embeddings via fp32 WMMA GEMM (memory-bound: keep full precision),
    //    A-slabs streamed with async global->LDS double buffering
    mn_embed_gemm<<<(Qn + 15) / 16, GEMM_THREADS, 0, stream>>>(
        query_images, backbone_w, backbone_b, q_emb, Qn, Din);
    mn_embed_gemm<<<(S + 15) / 16, GEMM_THREADS, 0, stream>>>(
        support_images, backbone_w, backbone_b, s_emb, S, Din);

    // 2) Normalize support embeddings
    mn_normalize_rows<<<S, EMB_D, 0, stream>>>(s_emb, s_norm);

    // 3) Cosine sims + softmax + label scatter
    mn_attn_logits<<<Qn, 128, 0, stream>>>(
        q_emb, s_norm, support_labels, out, S, n_way);
}